// Pooling_87101936762937
// MI455X (gfx1250) — compile-verified
//
#include <hip/hip_runtime.h>

// Segment mean pooling: out[b,:] = sum(x[ptr[b]:ptr[b+1],:]) / max(count,1)
// x: [524288, 128] f32, ptr: [1025] i32, out: [1024, 128] f32.
//
// HBM-bound (256 MiB @ 23.3 TB/s ~= 11.5 us floor). Strategy:
//  - 1 workgroup per segment, 8 waves (wave32) = one 16-col tile per wave.
//  - Row-sum on the fp32 matrix pipe: D = ones(16x4) x B(4x16) + D via
//    V_WMMA_F32_16X16X4_F32 (exact fp32 accumulate, 1024 MACs/instr, VALU free).
//  - 2x unrolled (8 rows/iter): 4 NT loads clause together before the first
//    s_wait_loadcnt -> double the per-wave outstanding VMEM.
//  - Non-temporal loads (array > 192 MB L2, each row read exactly once) +
//    global_prefetch_b8 a few iterations ahead.
//  - C/D VGPR0 on lanes 0..15 holds the column sums (M=0 row) -> one 64B store.

typedef __attribute__((ext_vector_type(2))) float v2f;
typedef __attribute__((ext_vector_type(8))) float v8f;

#define D_FEAT 128
#define N_GRAPHS 1024

__global__ __launch_bounds__(256)
void seg_mean_wmma_f32(const float* __restrict__ x,
                       const int* __restrict__ ptr,
                       float* __restrict__ out) {
  const int seg  = blockIdx.x;                  // one segment per block
  const int wave = threadIdx.x >> 5;            // 0..7 -> 16-column tile
  const int lane = threadIdx.x & 31;

  const int start = ptr[seg];
  const int end   = ptr[seg + 1];
  const int count = end - start;

  const int col  = wave * 16 + (lane & 15);     // this lane's feature column
  const int rsel = lane >> 4;                   // row parity within 4-row tile

  // A = ones(16x4): every D row becomes the sum of the 4 B rows.
  v2f ones; ones.x = 1.0f; ones.y = 1.0f;
  v8f acc = {};                                 // 16x16 f32 accumulator (8 VGPRs)

  int r = start;

  // Main loop, 8 rows per iteration (two 4-row WMMA tiles). All four NT
  // loads are issued back-to-back (one clause) before the first use.
  for (; r + 8 <= end; r += 8) {
    const float* p = x + (size_t)(r + rsel) * D_FEAT + col;
    v2f b0, b1;
    b0.x = __builtin_nontemporal_load(p);                 // rows r+0 / r+1
    b0.y = __builtin_nontemporal_load(p + 2 * D_FEAT);    // rows r+2 / r+3
    b1.x = __builtin_nontemporal_load(p + 4 * D_FEAT);    // rows r+4 / r+5
    b1.y = __builtin_nontemporal_load(p + 6 * D_FEAT);    // rows r+6 / r+7
    __builtin_prefetch(p + 32 * D_FEAT, 0, 1);            // global_prefetch_b8
    acc = __builtin_amdgcn_wmma_f32_16x16x4_f32(
        false, ones, false, b0, (short)0, acc, false, false);
    acc = __builtin_amdgcn_wmma_f32_16x16x4_f32(
        false, ones, false, b1, (short)0, acc, false, false);
  }

  // One remaining full 4-row tile (wave-uniform branch).
  if (r + 4 <= end) {
    const float* p = x + (size_t)(r + rsel) * D_FEAT + col;
    v2f b;
    b.x = __builtin_nontemporal_load(p);
    b.y = __builtin_nontemporal_load(p + 2 * D_FEAT);
    acc = __builtin_amdgcn_wmma_f32_16x16x4_f32(
        false, ones, false, b, (short)0, acc, false, false);
    r += 4;
  }

  // Tail (count % 4): zero-pad missing rows; the outer branch is wave-uniform
  // so EXEC is all-ones at the WMMA (ISA requirement) — the divergent part is
  // only the cndmask-selected loads, which re-converge before the WMMA.
  if (r < end) {
    const int r0 = r + rsel;
    const int r1 = r0 + 2;
    v2f b;
    b.x = (r0 < end) ? x[(size_t)r0 * D_FEAT + col] : 0.0f;
    b.y = (r1 < end) ? x[(size_t)r1 * D_FEAT + col] : 0.0f;
    acc = __builtin_amdgcn_wmma_f32_16x16x4_f32(
        false, ones, false, b, (short)0, acc, false, false);
  }

  // Mean (empty segment -> acc==0, inv over max(count,1) -> 0, matches ref).
  const float inv = 1.0f / (float)(count > 0 ? count : 1);

  // D layout: VGPR0, lanes 0..15 = row M=0, N=lane -> the 16 column sums.
  if (lane < 16) {
    out[(size_t)seg * D_FEAT + col] = acc[0] * inv;
  }
}

extern "C" void kernel_launch(void* const* d_in, const int* in_sizes, int n_in,
                              void* d_out, int out_size, void* d_ws, size_t ws_size,
                              hipStream_t stream) {
  const float* x   = (const float*)d_in[0];   // [N_NODES, D_FEAT] f32
  const int*   ptr = (const int*)d_in[1];     // [N_GRAPHS + 1] i32
  float*       out = (float*)d_out;           // [N_GRAPHS, D_FEAT] f32
  (void)in_sizes; (void)n_in; (void)out_size; (void)d_ws; (void)ws_size;

  seg_mean_wmma_f32<<<N_GRAPHS, 256, 0, stream>>>(x, ptr, out);
}